// MetaEdgePooling_56908316672642
// MI455X (gfx1250) — compile-verified
//
#include <hip/hip_runtime.h>
#include <stdint.h>

#define NN 25000      // nodes
#define NE 200000     // edges
#define FD 128        // features
#define SORT_N 262144 // next pow2 >= NE
#define SCAN_T 256

typedef __attribute__((ext_vector_type(2))) float v2f;
typedef __attribute__((ext_vector_type(8))) float v8f;

// order-preserving float <-> uint key (ascending uint == ascending float)
__device__ __forceinline__ unsigned fkey(float f) {
  unsigned u = __float_as_uint(f);
  return (u & 0x80000000u) ? ~u : (u | 0x80000000u);
}
__device__ __forceinline__ float fkeyinv(unsigned k) {
  unsigned u = (k & 0x80000000u) ? (k & 0x7fffffffu) : ~k;
  return __uint_as_float(u);
}

// ---------------- per-node scores: [NN,FD] x [FD,2] GEMM via f32 WMMA ----
// A(16x4) layout: lane = half*16 + (M%16); VGPR0/1 hold K = 2*half, 2*half+1.
// B(4x16): lane column = lane&15; only columns 0 (w_src) and 1 (w_dst) live.
// D(16x16): VGPR r -> M = r + 8*half, N = lane&15.
__global__ void __launch_bounds__(256)
node_scores_wmma(const float* __restrict__ x,
                 const float* __restrict__ w_src,
                 const float* __restrict__ w_dst,
                 float* __restrict__ s_src,
                 float* __restrict__ s_dst) {
  const int lane = threadIdx.x & 31;
  const int tile = blockIdx.x * (blockDim.x >> 5) + (threadIdx.x >> 5);
  const int half = lane >> 4;
  const int col  = lane & 15;
  // clamp instead of predicate: keeps every load uniform/unconditional;
  // duplicate rows are discarded by the guarded write below.
  const int arow = min(tile * 16 + col, NN - 1);
  const float* __restrict__ xrow = x + (size_t)arow * FD + 2 * half;

  // per-lane B-column blend factors (hoisted: no branches in the K loop)
  const float selS = (col == 0) ? 1.0f : 0.0f;
  const float selD = (col == 1) ? 1.0f : 0.0f;

  v8f acc = {};
#pragma unroll 4
  for (int k0 = 0; k0 < FD; k0 += 4) {
    const int k = k0 + 2 * half;
    v2f a  = *(const v2f*)(xrow + k0);        // x[arow][k..k+1], 8B aligned
    v2f ws = *(const v2f*)(w_src + k);
    v2f wd = *(const v2f*)(w_dst + k);
    v2f bm;
    bm.x = ws.x * selS + wd.x * selD;
    bm.y = ws.y * selS + wd.y * selD;
    acc = __builtin_amdgcn_wmma_f32_16x16x4_f32(false, a, false, bm,
                                                (short)0, acc, false, false);
  }
  if (col < 2) {
    float* outp = (col == 0) ? s_src : s_dst;
#pragma unroll
    for (int r = 0; r < 8; ++r) {
      int m = tile * 16 + half * 8 + r;
      if (m < NN) outp[m] = acc[r];
    }
  }
}

// ---------------- init: zero outputs / accumulators, pad sort keys ---------
__global__ void init_kernel(float* new_x, int* new_batch, unsigned* mkey,
                            float* denom, unsigned long long* keys) {
  int i = blockIdx.x * blockDim.x + threadIdx.x;
  int stride = gridDim.x * blockDim.x;
  for (int j = i; j < NN * FD; j += stride) new_x[j] = 0.0f;
  for (int j = i; j < NN; j += stride) {
    new_batch[j] = 0;
    mkey[j] = 0u;          // below every real score key
    denom[j] = 0.0f;
  }
  for (int j = i + NE; j < SORT_N; j += stride)
    keys[j] = 0xFFFFFFFFFFFFFFFFull;  // pad sorts after all real edges
}

// ---------------- raw edge score + segment max ----------------------------
__global__ void edge_raw_max(const int* __restrict__ src, const int* __restrict__ dst,
                             const float* __restrict__ s_src, const float* __restrict__ s_dst,
                             const float* __restrict__ bptr,
                             float* __restrict__ raw, unsigned* __restrict__ mkey) {
  int e = blockIdx.x * blockDim.x + threadIdx.x;
  if (e >= NE) return;
  float r = s_src[src[e]] + s_dst[dst[e]] + bptr[0];
  raw[e] = r;
  atomicMax(&mkey[dst[e]], fkey(r));
}

// ---------------- z = exp(raw - max) + segment sum ------------------------
__global__ void edge_z_denom(const int* __restrict__ dst, const float* __restrict__ raw,
                             const unsigned* __restrict__ mkey,
                             float* __restrict__ z, float* __restrict__ denom) {
  int e = blockIdx.x * blockDim.x + threadIdx.x;
  if (e >= NE) return;
  int d = dst[e];
  float zz = expf(raw[e] - fkeyinv(mkey[d]));
  z[e] = zz;
  atomicAdd(&denom[d], zz);
}

// ---------------- final score + 64-bit sort key ---------------------------
__global__ void edge_score_key(const int* __restrict__ dst, const float* __restrict__ z,
                               const float* __restrict__ denom,
                               float* __restrict__ escore,
                               unsigned long long* __restrict__ keys) {
  int e = blockIdx.x * blockDim.x + threadIdx.x;
  if (e >= NE) return;
  float es = z[e] / denom[dst[e]] + 0.5f;
  escore[e] = es;
  // ascending sort => descending score, stable (edge id) tie-break
  keys[e] = (((unsigned long long)(~fkey(es))) << 32) | (unsigned)e;
}

// ---------------- bitonic compare-exchange pass ---------------------------
__global__ void bitonic_step(unsigned long long* __restrict__ keys, int j, int k) {
  int i = blockIdx.x * blockDim.x + threadIdx.x;
  int p = i ^ j;
  if (p > i) {
    unsigned long long a = keys[i], b = keys[p];
    bool swap = ((i & k) == 0) ? (a > b) : (a < b);
    if (swap) { keys[i] = b; keys[p] = a; }
  }
}

// ---------------- sequential greedy contraction (LDS-resident state) ------
__global__ void __launch_bounds__(SCAN_T)
greedy_scan(const unsigned long long* __restrict__ keys,
            const int* __restrict__ src, const int* __restrict__ dst,
            const float* __restrict__ escore,
            int* __restrict__ cluster, float* __restrict__ cscore,
            int* __restrict__ members, int* __restrict__ numc_ws,
            int* __restrict__ numc_out) {
  __shared__ unsigned mbits[(NN + 31) / 32];     // matched bitset, 3.1 KB
  __shared__ int   ls[SCAN_T], lt[SCAN_T], lcl[SCAN_T];
  __shared__ float lsc[SCAN_T];
  const int tid = threadIdx.x;

  for (int w = tid; w < (NN + 31) / 32; w += SCAN_T) mbits[w] = 0u;
  __syncthreads();

  int cnt = 0;  // meaningful on thread 0 only
  for (int base = 0; base < NE; base += SCAN_T) {
    __builtin_prefetch(keys + base + SCAN_T, 0, 0);   // global_prefetch_b8
    int pos = base + tid;
    if (pos < NE) {
      unsigned long long kk = keys[pos];
      int id = (int)(unsigned)(kk & 0xFFFFFFFFull);
      ls[tid] = src[id];
      lt[tid] = dst[id];
      lsc[tid] = escore[id];
    }
    __syncthreads();
    if (tid == 0) {
      int lim = min(SCAN_T, NE - base);
      for (int t = 0; t < lim; ++t) {
        int s = ls[t], d = lt[t];
        unsigned bs = (mbits[s >> 5] >> (s & 31)) & 1u;
        unsigned bd = (mbits[d >> 5] >> (d & 31)) & 1u;
        if (!(bs | bd)) {
          mbits[s >> 5] |= 1u << (s & 31);
          mbits[d >> 5] |= 1u << (d & 31);
          lcl[t] = cnt++;
        } else {
          lcl[t] = -1;
        }
      }
    }
    __syncthreads();
    if (pos < NE && lcl[tid] >= 0) {
      int c = lcl[tid];
      int s = ls[tid], d = lt[tid];
      cluster[s] = c;
      cluster[d] = c;
      cscore[c] = lsc[tid];
      members[2 * c] = s;
      members[2 * c + 1] = (d == s) ? -1 : d;
    }
    __syncthreads();
  }
  if (tid == 0) {
    for (int n = 0; n < NN; ++n) {
      if (!((mbits[n >> 5] >> (n & 31)) & 1u)) {
        cluster[n] = cnt;
        cscore[cnt] = 1.0f;
        members[2 * cnt] = n;
        members[2 * cnt + 1] = -1;
        ++cnt;
      }
    }
    *numc_ws = cnt;
    *numc_out = cnt;
  }
}

// ---------------- new_x[c] = (x[m0] + x[m1]) * cscore[c] (no atomics) -----
__global__ void build_new_x(const float* __restrict__ x, const int* __restrict__ members,
                            const float* __restrict__ cscore, const int* __restrict__ numc,
                            float* __restrict__ new_x) {
  int c = blockIdx.x;
  if (c >= *numc) return;            // rows >= num_clusters stay zero
  int f = threadIdx.x;
  int m0 = members[2 * c], m1 = members[2 * c + 1];
  float v = x[m0 * FD + f];
  if (m1 >= 0) v += x[m1 * FD + f];
  new_x[c * FD + f] = v * cscore[c];
}

// ---------------- relabel edges, mark self-loops --------------------------
__global__ void build_new_ei(const int* __restrict__ src, const int* __restrict__ dst,
                             const int* __restrict__ cluster, int* __restrict__ out_ei) {
  int e = blockIdx.x * blockDim.x + threadIdx.x;
  if (e >= NE) return;
  int a = cluster[src[e]];
  int b = cluster[dst[e]];
  if (a == b) { a = -1; b = -1; }
  out_ei[e] = a;
  out_ei[NE + e] = b;
}

extern "C" void kernel_launch(void* const* d_in, const int* in_sizes, int n_in,
                              void* d_out, int out_size, void* d_ws, size_t ws_size,
                              hipStream_t stream) {
  (void)in_sizes; (void)n_in; (void)out_size; (void)ws_size;
  const float* x      = (const float*)d_in[0];
  const int*   ei     = (const int*)d_in[1];    // [2, NE]
  const float* w_src  = (const float*)d_in[3];
  const float* w_dst  = (const float*)d_in[4];
  const float* bptr   = (const float*)d_in[5];
  const int* src = ei;
  const int* dst = ei + NE;

  // output layout (flat, return order): new_x | new_ei | new_batch | numc
  float* new_x    = (float*)d_out;
  int*   new_ei   = (int*)d_out + (size_t)NN * FD;
  int*   new_bat  = new_ei + 2 * NE;
  int*   numc_out = new_bat + NN;

  // workspace carve (256B aligned slots)
  char* ws = (char*)d_ws;
  size_t off = 0;
  auto carve = [&](size_t bytes) -> void* {
    void* p = ws + off;
    off += (bytes + 255) & ~(size_t)255;
    return p;
  };
  float* s_src  = (float*)carve(sizeof(float) * NN);
  float* s_dst  = (float*)carve(sizeof(float) * NN);
  float* raw    = (float*)carve(sizeof(float) * NE);
  unsigned* mkey = (unsigned*)carve(sizeof(unsigned) * NN);
  float* denom  = (float*)carve(sizeof(float) * NN);
  float* zbuf   = (float*)carve(sizeof(float) * NE);
  float* escore = (float*)carve(sizeof(float) * NE);
  unsigned long long* keys = (unsigned long long*)carve(sizeof(unsigned long long) * SORT_N);
  int* cluster  = (int*)carve(sizeof(int) * NN);
  float* cscore = (float*)carve(sizeof(float) * NN);
  int* members  = (int*)carve(sizeof(int) * 2 * NN);
  int* numc_ws  = (int*)carve(sizeof(int));

  init_kernel<<<512, 256, 0, stream>>>(new_x, new_bat, mkey, denom, keys);

  {
    const int tiles = (NN + 15) / 16;       // 1563 wave-tiles
    const int wpb = 8;                       // 256 threads = 8 waves
    node_scores_wmma<<<(tiles + wpb - 1) / wpb, 256, 0, stream>>>(
        x, w_src, w_dst, s_src, s_dst);
  }

  const int eb = (NE + 255) / 256;
  edge_raw_max<<<eb, 256, 0, stream>>>(src, dst, s_src, s_dst, bptr, raw, mkey);
  edge_z_denom<<<eb, 256, 0, stream>>>(dst, raw, mkey, zbuf, denom);
  edge_score_key<<<eb, 256, 0, stream>>>(dst, zbuf, denom, escore, keys);

  for (int k = 2; k <= SORT_N; k <<= 1)
    for (int j = k >> 1; j >= 1; j >>= 1)
      bitonic_step<<<SORT_N / 256, 256, 0, stream>>>(keys, j, k);

  greedy_scan<<<1, SCAN_T, 0, stream>>>(keys, src, dst, escore, cluster, cscore,
                                        members, numc_ws, numc_out);

  build_new_x<<<NN, FD, 0, stream>>>(x, members, cscore, numc_ws, new_x);
  build_new_ei<<<eb, 256, 0, stream>>>(src, dst, cluster, new_ei);
}